// ShiftedWindowMSA_51539607991
// MI455X (gfx1250) — compile-verified
//
#include <hip/hip_runtime.h>
#include <hip/hip_bf16.h>

typedef __attribute__((ext_vector_type(2))) float v2f;
typedef __attribute__((ext_vector_type(8))) float v8f;

#define BDIM   128
#define GRID   56
#define HWTOK  3136
#define EDIM   96
#define NHEAD  3
#define EH     32
#define WS     7
#define NWIN   8          // 56/7 per axis
#define WTOK   49         // 7*7
#define MROWS  (BDIM*HWTOK)   // 401408
#define N1     (3*EDIM)       // 288

// ---------------------------------------------------------------------------
// Generic fp32 WMMA GEMM:  C[M,N] = A[M,K] * Bw[N,K]^T + bias[N]
// One wave computes one 16x16 C tile with V_WMMA_F32_16X16X4_F32.
// A-layout per ISA: lane L holds A[M = tile_m*16 + L%16][K = k + (L/16)*2 + {0,1}]
// B-layout:         lane L holds B[K = k + (L/16)*2 + {0,1}][N = tile_n*16 + L%16]
//                   = Bw[tile_n*16 + L%16][k + (L/16)*2 + {0,1}]  (contiguous float2)
// C-layout: VGPR j: lanes 0-15 -> row j, lanes 16-31 -> row j+8, col = L%16.
// ---------------------------------------------------------------------------
__global__ __launch_bounds__(256) void
wmma_gemm_bias(const float* __restrict__ A, const float* __restrict__ Bw,
               const float* __restrict__ bias, float* __restrict__ C,
               long M, int N, int K)
{
    const int lane = threadIdx.x & 31;
    const int wave = threadIdx.x >> 5;
    const int l16  = lane & 15;
    const int lh   = lane >> 4;

    const long tile   = (long)blockIdx.x * 8 + wave;
    const int  ntiles = N >> 4;
    const long mt     = tile / ntiles;
    const int  nt     = (int)(tile % ntiles);
    if (mt >= (M >> 4)) return;          // wave-uniform; EXEC stays all-ones inside

    const float* arow = A  + ((mt << 4) + l16) * (long)K;
    const float* brow = Bw + ((long)((nt << 4) + l16)) * (long)K;

    v8f acc = {};
    for (int k = 0; k < K; k += 4) {
        const int ka = k + (lh << 1);
        v2f av = *(const v2f*)(arow + ka);
        v2f bv = *(const v2f*)(brow + ka);
        acc = __builtin_amdgcn_wmma_f32_16x16x4_f32(false, av, false, bv,
                                                    (short)0, acc, false, false);
    }

    const float bz = bias[(nt << 4) + l16];
    float* cptr = C + (nt << 4) + l16;
#pragma unroll
    for (int j = 0; j < 8; ++j) {
        const long row = (mt << 4) + j + (lh << 3);
        cptr[row * (long)N] = acc[j] + bz;
    }
}

// ---------------------------------------------------------------------------
// Shifted-window attention, one workgroup (128 thr = 4 waves) per (b, head, window).
// Q,K,V staged to LDS (64x32, zero padded); att = Q*K^T padded to 64x64 via WMMA;
// scale + swin masks; shuffle row-softmax in C-layout; P through LDS; O = P*V.
// ---------------------------------------------------------------------------
__global__ __launch_bounds__(128) void
swin_attention(const float* __restrict__ qkv, float* __restrict__ ofull)
{
    __shared__ float sQ[64][32];
    __shared__ float sK[64][32];
    __shared__ float sV[64][32];
    __shared__ float sP[64][64];

    const int blk  = blockIdx.x;
    const int win  = blk & 63;          // 8x8 windows
    const int head = (blk >> 6) % NHEAD;
    const int bb   = blk / (64 * NHEAD);
    const int winy = win >> 3;
    const int winx = win & 7;

    const int tid  = threadIdx.x;
    const int lane = tid & 31;
    const int wave = tid >> 5;
    const int l16  = lane & 15;
    const int lh   = lane >> 4;

    // ---- gather Q/K/V with the -4 cyclic shift; zero-pad tokens 49..63 ----
    for (int idx = tid; idx < 64 * 32; idx += 128) {
        const int t  = idx >> 5;
        const int eh = idx & 31;
        float q = 0.f, k = 0.f, v = 0.f;
        if (t < WTOK) {
            const int m1 = t / WS, m2 = t % WS;
            int h = winy * WS + m1 + 4; if (h >= GRID) h -= GRID;
            int w = winx * WS + m2 + 4; if (w >= GRID) w -= GRID;
            const long base = ((long)(bb * HWTOK + h * GRID + w)) * N1
                            + (long)(head * EH + eh) * 3;
            q = qkv[base + 0];
            k = qkv[base + 1];
            v = qkv[base + 2];
        }
        sQ[t][eh] = q; sK[t][eh] = k; sV[t][eh] = v;
    }
    __syncthreads();

    // ---- att strip: this wave owns rows [wave*16, wave*16+16), all 64 cols ----
    v8f acc[4] = {};
    {
        const int mrow = (wave << 4) + l16;
        for (int ks = 0; ks < EH; ks += 4) {
            const int ka = ks + (lh << 1);
            v2f av = *(const v2f*)&sQ[mrow][ka];
#pragma unroll
            for (int t = 0; t < 4; ++t) {
                v2f bv = *(const v2f*)&sK[(t << 4) + l16][ka];
                acc[t] = __builtin_amdgcn_wmma_f32_16x16x4_f32(false, av, false, bv,
                                                               (short)0, acc[t], false, false);
            }
        }
    }

    // ---- scale + masks (padded keys, swin row/col masks) ----
    const bool lastRow = (winy == NWIN - 1);
    const bool lastCol = (winx == NWIN - 1);
    const float scale = 0.17677669529663687f;   // 1/sqrt(32)
#pragma unroll
    for (int t = 0; t < 4; ++t) {
#pragma unroll
        for (int j = 0; j < 8; ++j) {
            const int row = (wave << 4) + j + (lh << 3);
            const int col = (t << 4) + l16;
            float v = acc[t][j] * scale;
            bool dead = (col >= WTOK);
            if (lastRow && ((row >= 28) != (col >= 28)))           dead = true;
            if (lastCol && (((row % WS) >= 4) != ((col % WS) >= 4))) dead = true;
            acc[t][j] = dead ? -1e30f : v;
        }
    }

    // ---- row softmax (reduce across 4 col-tiles + 16 lanes of this half) ----
#pragma unroll
    for (int j = 0; j < 8; ++j) {
        float mx = fmaxf(fmaxf(acc[0][j], acc[1][j]), fmaxf(acc[2][j], acc[3][j]));
        mx = fmaxf(mx, __shfl_xor(mx, 1));
        mx = fmaxf(mx, __shfl_xor(mx, 2));
        mx = fmaxf(mx, __shfl_xor(mx, 4));
        mx = fmaxf(mx, __shfl_xor(mx, 8));
        float s = 0.f;
#pragma unroll
        for (int t = 0; t < 4; ++t) {
            float e = __expf(acc[t][j] - mx);
            acc[t][j] = e;
            s += e;
        }
        s += __shfl_xor(s, 1);
        s += __shfl_xor(s, 2);
        s += __shfl_xor(s, 4);
        s += __shfl_xor(s, 8);
        const float inv = 1.0f / s;
#pragma unroll
        for (int t = 0; t < 4; ++t) acc[t][j] *= inv;
    }

    // ---- stage P to LDS to convert C-layout -> A-layout ----
#pragma unroll
    for (int t = 0; t < 4; ++t)
#pragma unroll
        for (int j = 0; j < 8; ++j)
            sP[(wave << 4) + j + (lh << 3)][(t << 4) + l16] = acc[t][j];
    __syncthreads();

    // ---- O = P * V   (M=64 rows, K=64 keys, N=32 channels) ----
    v8f oacc[2] = {};
    {
        const int mrow = (wave << 4) + l16;
        for (int ks = 0; ks < 64; ks += 4) {
            const int ka = ks + (lh << 1);
            v2f av = *(const v2f*)&sP[mrow][ka];
#pragma unroll
            for (int t = 0; t < 2; ++t) {
                v2f bv;
                bv.x = sV[ka + 0][(t << 4) + l16];
                bv.y = sV[ka + 1][(t << 4) + l16];
                oacc[t] = __builtin_amdgcn_wmma_f32_16x16x4_f32(false, av, false, bv,
                                                                (short)0, oacc[t], false, false);
            }
        }
    }

    // ---- scatter O with the +3 reverse shift ----
#pragma unroll
    for (int t = 0; t < 2; ++t) {
#pragma unroll
        for (int j = 0; j < 8; ++j) {
            const int row = (wave << 4) + j + (lh << 3);
            if (row < WTOK) {
                const int eh = (t << 4) + l16;
                const int m1 = row / WS, m2 = row % WS;
                int ho = winy * WS + m1 + 3; if (ho >= GRID) ho -= GRID;
                int wo = winx * WS + m2 + 3; if (wo >= GRID) wo -= GRID;
                ofull[((long)(bb * HWTOK + ho * GRID + wo)) * EDIM + head * EH + eh] =
                    oacc[t][j];
            }
        }
    }
}

// ---------------------------------------------------------------------------
extern "C" void kernel_launch(void* const* d_in, const int* in_sizes, int n_in,
                              void* d_out, int out_size, void* d_ws, size_t ws_size,
                              hipStream_t stream)
{
    const float* x  = (const float*)d_in[0];
    const float* w1 = (const float*)d_in[1];
    const float* b1 = (const float*)d_in[2];
    const float* w2 = (const float*)d_in[3];
    const float* b2 = (const float*)d_in[4];
    float* out = (float*)d_out;

    float* qkv   = (float*)d_ws;                        // [MROWS, 288]
    float* ofull = qkv + (size_t)MROWS * N1;            // [MROWS, 96]

    // GEMM 1: qkv = x @ w1^T + b1   (M=401408, N=288, K=96)
    {
        const long tiles = (long)(MROWS / 16) * (N1 / 16);   // 451584
        wmma_gemm_bias<<<(int)(tiles / 8), 256, 0, stream>>>(
            x, w1, b1, qkv, (long)MROWS, N1, EDIM);
    }

    // Attention: one block per (b, head, window)
    {
        const int blocks = BDIM * NHEAD * 64;                // 24576
        swin_attention<<<blocks, 128, 0, stream>>>(qkv, ofull);
    }

    // GEMM 2: out = ofull @ w2^T + b2   (M=401408, N=96, K=96)
    {
        const long tiles = (long)(MROWS / 16) * (EDIM / 16); // 150528
        wmma_gemm_bias<<<(int)(tiles / 8), 256, 0, stream>>>(
            ofull, w2, b2, out, (long)MROWS, EDIM, EDIM);
    }
}